// Network_12137577578757
// MI455X (gfx1250) — compile-verified
//
#include <hip/hip_runtime.h>
#include <hip/hip_bf16.h>
#include <cstdint>

typedef __attribute__((ext_vector_type(2))) float v2f;
typedef __attribute__((ext_vector_type(8))) float v8f;

#define BN_EPS 1e-5f

// ---------------------------------------------------------------------------
// 1. xyz normalization: per (batch, channel) mean & var (ddof=1) over N, then
//    xyz = (xyz - mean) / (var + 1e-8)   (reference divides by VAR, not std)
// ---------------------------------------------------------------------------
__global__ __launch_bounds__(256)
void normalize_kernel(const float* __restrict__ in, float* __restrict__ xyzN, int N) {
  int b = blockIdx.x, tid = threadIdx.x;
  const float* src = in + (size_t)b * N * 6;
  float s0 = 0, s1 = 0, s2 = 0, q0 = 0, q1 = 0, q2 = 0;
  for (int p = tid; p < N; p += 256) {
    float x = src[p * 6 + 0], y = src[p * 6 + 1], z = src[p * 6 + 2];
    s0 += x; s1 += y; s2 += z;
    q0 += x * x; q1 += y * y; q2 += z * z;
  }
  __shared__ float red[6 * 256];
  red[0 * 256 + tid] = s0; red[1 * 256 + tid] = s1; red[2 * 256 + tid] = s2;
  red[3 * 256 + tid] = q0; red[4 * 256 + tid] = q1; red[5 * 256 + tid] = q2;
  __syncthreads();
  for (int st = 128; st > 0; st >>= 1) {
    if (tid < st)
      for (int j = 0; j < 6; ++j) red[j * 256 + tid] += red[j * 256 + tid + st];
    __syncthreads();
  }
  __shared__ float mean[3], scal[3];
  if (tid < 3) {
    float su = red[tid * 256], sq = red[(3 + tid) * 256];
    float mu = su / (float)N;
    float var = (sq - su * mu) / (float)(N - 1);
    mean[tid] = mu;
    scal[tid] = 1.0f / (var + 1e-8f);
  }
  __syncthreads();
  for (int p = tid; p < N; p += 256) {
    size_t o = ((size_t)b * N + p) * 3;
    xyzN[o + 0] = (src[p * 6 + 0] - mean[0]) * scal[0];
    xyzN[o + 1] = (src[p * 6 + 1] - mean[1]) * scal[1];
    xyzN[o + 2] = (src[p * 6 + 2] - mean[2]) * scal[2];
  }
}

// ---------------------------------------------------------------------------
// 2. Farthest-point sampling. One workgroup per batch, 256 threads, strided
//    point ownership (<=32 pts/thread). Argmax with first-index tiebreak to
//    match jnp.argmax. Also gathers new_xyz at the end.
// ---------------------------------------------------------------------------
__global__ __launch_bounds__(256)
void fps_kernel(const float* __restrict__ xyz, int N, int npoint,
                int* __restrict__ idx, float* __restrict__ newxyz) {
  int b = blockIdx.x, tid = threadIdx.x;
  const float* X = xyz + (size_t)b * N * 3;
  float d[32];
  int cnt = 0;
  for (int p = tid; p < N; p += 256) d[cnt++] = 1e10f;

  __shared__ float rv[256];
  __shared__ int   ri[256];
  __shared__ float lp[3];
  __shared__ int   sel[1024];
  if (tid == 0) sel[0] = 0;
  int last = 0;
  for (int it = 1; it < npoint; ++it) {
    if (tid == 0) { lp[0] = X[last*3]; lp[1] = X[last*3+1]; lp[2] = X[last*3+2]; }
    __syncthreads();
    float lx = lp[0], ly = lp[1], lz = lp[2];
    float bestv = -1.0f; int besti = N;
    int c = 0;
    for (int p = tid; p < N; p += 256, ++c) {
      float dx = X[p*3] - lx, dy = X[p*3+1] - ly, dz = X[p*3+2] - lz;
      float dd = dx*dx + dy*dy + dz*dz;
      float nd = fminf(d[c], dd);
      d[c] = nd;
      if (nd > bestv) { bestv = nd; besti = p; }
    }
    rv[tid] = bestv; ri[tid] = besti;
    __syncthreads();
    for (int st = 128; st > 0; st >>= 1) {
      if (tid < st) {
        float ov = rv[tid + st]; int oi = ri[tid + st];
        if (ov > rv[tid] || (ov == rv[tid] && oi < ri[tid])) { rv[tid] = ov; ri[tid] = oi; }
      }
      __syncthreads();
    }
    last = ri[0];
    if (tid == 0) sel[it] = last;
    __syncthreads();
  }
  for (int j = tid; j < npoint; j += 256) {
    int k = sel[j];
    idx[(size_t)b * npoint + j] = k;
    size_t o = ((size_t)b * npoint + j) * 3;
    newxyz[o + 0] = X[k*3]; newxyz[o + 1] = X[k*3+1]; newxyz[o + 2] = X[k*3+2];
  }
}

// ---------------------------------------------------------------------------
// 3. Ball query + grouping. One wave (32 lanes) per centroid. Selects first
//    32 in-radius indices in ascending order via ballot compaction (wave32:
//    one 32-point chunk per step), pads with slot 0, writes padded rows
//    [gxyz(3) | feats(C) | zeros..Cpad].
// ---------------------------------------------------------------------------
__global__ __launch_bounds__(256)
void group_kernel(const float* __restrict__ xyz, const float* __restrict__ feats,
                  const float* __restrict__ newxyz, int N, int P, int C, int Cpad,
                  float r2, int nwaves, float* __restrict__ out) {
  int wid  = (int)((blockIdx.x * blockDim.x + threadIdx.x) >> 5);
  int lane = threadIdx.x & 31;
  int w    = threadIdx.x >> 5;
  __shared__ int sel[8][32];
  bool active = wid < nwaves;
  int b = 0, p = 0, taken = 0;
  float cx = 0, cy = 0, cz = 0;
  const float* X = xyz;
  if (active) {
    b = wid / P; p = wid % P;
    X = xyz + (size_t)b * N * 3;
    size_t co = ((size_t)b * P + p) * 3;
    cx = newxyz[co]; cy = newxyz[co + 1]; cz = newxyz[co + 2];
    for (int base = 0; base < N && taken < 32; base += 32) {
      int pi = base + lane;
      bool ok = false;
      if (pi < N) {
        float dx = X[pi*3] - cx, dy = X[pi*3+1] - cy, dz = X[pi*3+2] - cz;
        ok = (dx*dx + dy*dy + dz*dz) < r2;
      }
      unsigned m = (unsigned)__ballot(ok);
      int pre = __popc(m & ((1u << lane) - 1u));
      if (ok && (taken + pre) < 32) sel[w][taken + pre] = pi;
      taken += __popc(m);
      if (taken > 32) taken = 32;
    }
    if (taken == 0) { if (lane == 0) sel[w][0] = 0; taken = 1; }
  }
  __syncthreads();
  if (active) {
    int gi = (lane < taken) ? sel[w][lane] : sel[w][0];
    size_t row = (((size_t)b * P + p) * 32 + lane) * (size_t)Cpad;
    out[row + 0] = X[gi*3]   - cx;
    out[row + 1] = X[gi*3+1] - cy;
    out[row + 2] = X[gi*3+2] - cz;
    const float* F = feats + ((size_t)b * N + gi) * C;
    for (int c = 0; c < C; ++c) out[row + 3 + c] = F[c];
    for (int c = 3 + C; c < Cpad; ++c) out[row + c] = 0.0f;
  }
}

// ---------------------------------------------------------------------------
// 4. concat(xyz, feats) for the global SA module, K-padded.
// ---------------------------------------------------------------------------
__global__ __launch_bounds__(256)
void concat_kernel(const float* __restrict__ xyz, const float* __restrict__ feats,
                   int rows, int C, int Cpad, float* __restrict__ out) {
  int r = blockIdx.x * blockDim.x + threadIdx.x;
  if (r >= rows) return;
  float* o = out + (size_t)r * Cpad;
  o[0] = xyz[r*3]; o[1] = xyz[r*3+1]; o[2] = xyz[r*3+2];
  const float* F = feats + (size_t)r * C;
  for (int c = 0; c < C; ++c) o[3 + c] = F[c];
  for (int c = 3 + C; c < Cpad; ++c) o[c] = 0.0f;
}

// ---------------------------------------------------------------------------
// 5. Zero-pad weights (cout x Kact) -> (cout x Kpad) so the GEMM K-loop is
//    completely unconditional.
// ---------------------------------------------------------------------------
__global__ __launch_bounds__(256)
void wpad_kernel(const float* __restrict__ W, int Kact, int Kpad, int Nc,
                 float* __restrict__ Wp) {
  int i = blockIdx.x * blockDim.x + threadIdx.x;
  if (i >= Nc * Kpad) return;
  int o = i / Kpad, k = i % Kpad;
  Wp[i] = (k < Kact) ? W[o * Kact + k] : 0.0f;
}

// ---------------------------------------------------------------------------
// 6. WMMA GEMM: out = relu(bn(X @ W^T)), optional fused max-pool over `pool`
//    consecutive rows via integer atomicMax (valid: ReLU output >= 0).
//    One wave per 16x16 tile, V_WMMA_F32_16X16X4_F32 over K chunks of 4.
//    Preconditions: M % 16 == 0, Kpad % 4 == 0, X is M x Kpad, W is Nc x Kpad
//    (both zero-padded) -> all fragment loads are unconditional b64 vectors,
//    EXEC stays all-ones through every WMMA.
// ---------------------------------------------------------------------------
__global__ __launch_bounds__(256)
void gemm_bn_relu_pool(const float* __restrict__ X, const float* __restrict__ W,
                       const float* __restrict__ gg, const float* __restrict__ bb,
                       const float* __restrict__ mm, const float* __restrict__ vv,
                       int M, int Kpad, int Nc, int pool, int Mstore,
                       float* __restrict__ out) {
  int wave = (int)((blockIdx.x * blockDim.x + threadIdx.x) >> 5);
  int lane = threadIdx.x & 31;
  int Mt = M >> 4;
  int Nt = Nc >> 4;
  if (wave >= Mt * Nt) return;              // wave-uniform exit
  int tm = wave % Mt, tn = wave / Mt;
  int half = lane >> 4, r = lane & 15;

  // A frag: lane(0-15)=row r holds K = k0+{0,1}; lane(16-31) holds K = k0+{2,3}
  const float* ap = X + (size_t)(tm * 16 + r) * Kpad + 2 * half;
  // B frag (W^T chunk): same K split per half-wave, column = tn*16 + r
  const float* bp = W + (size_t)(tn * 16 + r) * Kpad + 2 * half;

  v8f acc = {};
#pragma unroll 4
  for (int k0 = 0; k0 < Kpad; k0 += 4) {
    v2f a  = *(const v2f*)(ap + k0);        // global_load_b64, 8B-aligned
    v2f bf = *(const v2f*)(bp + k0);        // global_load_b64, 8B-aligned
    acc = __builtin_amdgcn_wmma_f32_16x16x4_f32(
        /*neg_a=*/false, a, /*neg_b=*/false, bf,
        /*c_mod=*/(short)0, acc, /*reuse_a=*/false, /*reuse_b=*/false);
  }
  // BN folded to a*x + c per output column (lane's column is fixed: tn*16+r)
  int o = tn * 16 + r;
  float aco = gg[o] * rsqrtf(vv[o] + BN_EPS);
  float cco = bb[o] - mm[o] * aco;
#pragma unroll
  for (int i = 0; i < 8; ++i) {
    int row = tm * 16 + i + 8 * half;       // C/D layout: VGPR i, halves M / M+8
    float val = fmaxf(acc[i] * aco + cco, 0.0f);
    if (row < Mstore) {
      if (pool > 1) {
        atomicMax((int*)&out[(size_t)(row / pool) * Nc + o], __float_as_int(val));
      } else {
        out[(size_t)row * Nc + o] = val;
      }
    }
  }
}

// ---------------------------------------------------------------------------
// host orchestration
// ---------------------------------------------------------------------------
struct Lay { const float *W, *g, *b, *m, *v; int cout, cin; };

extern "C" void kernel_launch(void* const* d_in, const int* in_sizes, int n_in,
                              void* d_out, int out_size, void* d_ws, size_t ws_size,
                              hipStream_t stream) {
  const int B = 8, N0 = 8192;
  const float* batch = (const float*)d_in[0];

  // Param order: insertion order (batch; sa[4][3]{W,g,b,m,v}; fc[3]{...}).
  // Fallback: sorted pytree order (batch; fc first; {W,b,g,m,v}) detected by
  // in_sizes[1] == 512*1024 (fc0.W) vs 288 (sa0l0.W).
  bool fc_first = (n_in > 1 && in_sizes[1] == 512 * 1024);
  auto mk = [&](int base, int cout, int cin) -> Lay {
    Lay L;
    L.W = (const float*)d_in[base + 0];
    if (fc_first) { L.b = (const float*)d_in[base + 1]; L.g = (const float*)d_in[base + 2]; }
    else          { L.g = (const float*)d_in[base + 1]; L.b = (const float*)d_in[base + 2]; }
    L.m = (const float*)d_in[base + 3];
    L.v = (const float*)d_in[base + 4];
    L.cout = cout; L.cin = cin;
    return L;
  };
  int saB = fc_first ? 16 : 1;
  int fcB = fc_first ? 1  : 61;
  Lay sa[4][3] = {
    { mk(saB +  0, 32,   9), mk(saB +  5, 32,  32), mk(saB + 10,   64,  32) },
    { mk(saB + 15, 64,  67), mk(saB + 20, 64,  64), mk(saB + 25,  128,  64) },
    { mk(saB + 30, 128, 131), mk(saB + 35, 128, 128), mk(saB + 40, 256, 128) },
    { mk(saB + 45, 256, 259), mk(saB + 50, 512, 256), mk(saB + 55, 1024, 512) },
  };
  Lay fc[3] = { mk(fcB + 0, 512, 1024), mk(fcB + 5, 256, 512), mk(fcB + 10, 256, 256) };

  // workspace carve (float units)
  float* ws     = (float*)d_ws;
  float* xyzN   = ws;                          // 8*8192*3      = 196608
  int*   fidx   = (int*)(ws + 196608);         // 8*1024 ints   =   8192
  float* nx1    = ws + 196608 + 8192;          // 8*1024*3      =  24576
  float* f1     = nx1 + 24576;                 // 8*1024*64     = 524288
  float* nx2    = f1 + 524288;                 // 8*256*3       =   6144
  float* f2     = nx2 + 6144;                  // 8*256*128     = 262144
  float* nx3    = f2 + 262144;                 // 8*64*3        =   1536
  float* f3     = nx3 + 1536;                  // 8*64*256      = 131072
  float* pooled = f3 + 131072;                 // 16*1024       =  16384  (rows 8..15 stay 0)
  float* fcA    = pooled + 16384;              // 16*512        =   8192
  float* fcB2   = fcA + 8192;                  // 16*256        =   4096
  float* wpad   = fcB2 + 4096;                 // padded weights, <= 256*260 = 66560
  float* bufA   = wpad + 67584;                // 8*1024*32*32  = 8388608
  float* bufB   = bufA + 8388608;              // 8388608

  // zero pooled targets (atomicMax accumulators); pooled includes pad rows
  hipMemsetAsync(f1,     0, 524288 * sizeof(float), stream);
  hipMemsetAsync(f2,     0, 262144 * sizeof(float), stream);
  hipMemsetAsync(f3,     0, 131072 * sizeof(float), stream);
  hipMemsetAsync(pooled, 0,  16384 * sizeof(float), stream);

  auto gemm = [&](const float* X, const float* W, const Lay& L, int M, int Kpad,
                  int pool, int Mstore, float* out) {
    int waves = (M / 16) * (L.cout / 16);
    int blocks = (waves + 7) / 8;
    gemm_bn_relu_pool<<<blocks, 256, 0, stream>>>(X, W, L.g, L.b, L.m, L.v,
                                                  M, Kpad, L.cout, pool, Mstore, out);
  };
  auto wprep = [&](const Lay& L, int Kpad) {
    int total = L.cout * Kpad;
    wpad_kernel<<<(total + 255) / 256, 256, 0, stream>>>(L.W, L.cin, Kpad, L.cout, wpad);
  };

  normalize_kernel<<<B, 256, 0, stream>>>(batch, xyzN, N0);

  // ---- SA0: N=8192 -> P=1024, r=0.1, C=6 (Cpad 12) ----
  fps_kernel<<<B, 256, 0, stream>>>(xyzN, 8192, 1024, fidx, nx1);
  group_kernel<<<(B * 1024) / 8, 256, 0, stream>>>(xyzN, batch, nx1, 8192, 1024, 6, 12,
                                                   0.01f, B * 1024, bufA);
  wprep(sa[0][0], 12);
  gemm(bufA, wpad,       sa[0][0], 262144, 12, 1, 262144, bufB);
  gemm(bufB, sa[0][1].W, sa[0][1], 262144, 32, 1, 262144, bufA);
  gemm(bufA, sa[0][2].W, sa[0][2], 262144, 32, 32, 262144, f1);

  // ---- SA1: N=1024 -> P=256, r=0.2, C=64 (Cpad 68) ----
  fps_kernel<<<B, 256, 0, stream>>>(nx1, 1024, 256, fidx, nx2);
  group_kernel<<<(B * 256) / 8, 256, 0, stream>>>(nx1, f1, nx2, 1024, 256, 64, 68,
                                                  0.04f, B * 256, bufA);
  wprep(sa[1][0], 68);
  gemm(bufA, wpad,       sa[1][0], 65536, 68, 1, 65536, bufB);
  gemm(bufB, sa[1][1].W, sa[1][1], 65536, 64, 1, 65536, bufA);
  gemm(bufA, sa[1][2].W, sa[1][2], 65536, 64, 32, 65536, f2);

  // ---- SA2: N=256 -> P=64, r=0.4, C=128 (Cpad 132) ----
  fps_kernel<<<B, 256, 0, stream>>>(nx2, 256, 64, fidx, nx3);
  group_kernel<<<(B * 64) / 8, 256, 0, stream>>>(nx2, f2, nx3, 256, 64, 128, 132,
                                                 0.16f, B * 64, bufA);
  wprep(sa[2][0], 132);
  gemm(bufA, wpad,       sa[2][0], 16384, 132, 1, 16384, bufB);
  gemm(bufB, sa[2][1].W, sa[2][1], 16384, 128, 1, 16384, bufA);
  gemm(bufA, sa[2][2].W, sa[2][2], 16384, 128, 32, 16384, f3);

  // ---- SA3 (global): rows = 8*64, C=256 (Cpad 260), pool over 64 ----
  concat_kernel<<<2, 256, 0, stream>>>(nx3, f3, 512, 256, 260, bufA);
  wprep(sa[3][0], 260);
  gemm(bufA, wpad,       sa[3][0], 512, 260, 1, 512, bufB);
  gemm(bufB, sa[3][1].W, sa[3][1], 512, 256, 1, 512, bufA);
  gemm(bufA, sa[3][2].W, sa[3][2], 512, 512, 64, 512, pooled);

  // ---- FC head: rows padded 8 -> 16 (pad rows are zero / don't reach d_out)
  gemm(pooled, fc[0].W, fc[0], 16, 1024, 1, 16, fcA);
  gemm(fcA,    fc[1].W, fc[1], 16,  512, 1, 16, fcB2);
  gemm(fcB2,   fc[2].W, fc[2], 16,  256, 1,  8, (float*)d_out);

  (void)out_size; (void)ws_size; (void)n_in;
}